// DenseAtt_57157424775846
// MI455X (gfx1250) — compile-verified
//
#include <hip/hip_runtime.h>

// Problem constants (match reference: N=10000, F=256)
#define NR 10000
#define FDIM 256

typedef __attribute__((ext_vector_type(2))) float v2f;
typedef __attribute__((ext_vector_type(4))) float v4f;
typedef __attribute__((ext_vector_type(8))) float v8f;

// ---------------------------------------------------------------------------
// Kernel A: s = x @ [w1 w2] via V_WMMA_F32_16X16X4_F32, one wave per 16 rows.
//   A (16x4 f32): lane l holds row M = l&15; VGPR v at half h=(l>>4) is K=v+2h.
//   B (4x16 f32): lane l holds col N = l&15; VGPR v at half h is K=v+2h.
//   D (16x16 f32): VGPR r, lanes 0-15 -> M=r, lanes 16-31 -> M=r+8; N=lane&15.
// Columns 0/1 of B carry w1/w2; remaining columns are zeroed so only D[:,0..1]
// is meaningful. Lanes {0,16} emit p[i]=exp(-(s1[i]+b)); lanes {1,17} emit
// q[j]=exp(-s2[j]) into the workspace.
// ---------------------------------------------------------------------------
__global__ __launch_bounds__(32) void denseatt_matvec_wmma(
    const float* __restrict__ x,   // (NR, FDIM)
    const float* __restrict__ w,   // (1, 2*FDIM) flat: w1 | w2
    const float* __restrict__ bvec,// (1,)
    float* __restrict__ pq)        // ws: [0,NR) = p, [NR,2NR) = q
{
    const int lane = threadIdx.x;        // 0..31, full wave active (WMMA needs EXEC all-1)
    const int hi   = lane >> 4;          // half-wave select
    const int n    = lane & 15;          // M for A-loads, N for B/D
    const int r0   = blockIdx.x * 16;    // 625 blocks * 16 rows = 10000 exactly

    const float* xrow  = x + (size_t)(r0 + n) * FDIM;
    const int    wcol  = (n < 2) ? n : 0;            // safe address for all lanes
    const float* wbase = w + wcol * FDIM;
    const float  bmask = (n < 2) ? 1.0f : 0.0f;      // zero out columns 2..15

    v8f c = {0.f, 0.f, 0.f, 0.f, 0.f, 0.f, 0.f, 0.f};

#pragma unroll 4
    for (int k = 0; k < FDIM; k += 4) {
        const int kk = k + 2 * hi;                   // this half-wave's K pair
        v2f a  = *(const v2f*)(xrow + kk);           // aligned b64 load
        v2f wv = *(const v2f*)(wbase + kk);
        v2f bm;
        bm.x = bmask * wv.x;
        bm.y = bmask * wv.y;
        c = __builtin_amdgcn_wmma_f32_16x16x4_f32(
                /*neg_a=*/false, a, /*neg_b=*/false, bm,
                /*c_mod=*/(short)0, c, /*reuse_a=*/false, /*reuse_b=*/false);
    }

    if (n < 2) {
        const float b0 = bvec[0];
#pragma unroll
        for (int r = 0; r < 8; ++r) {
            const int   row = r0 + r + 8 * hi;       // D row M = r + 8*hi
            const float v   = c[r];                  // D[M][n]
            if (n == 0) pq[row]      = __expf(-(v + b0));  // p[i]
            else        pq[NR + row] = __expf(-v);         // q[j]
        }
    }
}

// ---------------------------------------------------------------------------
// Kernel B: att[i,j] = 1 / (1 + p[i]*q[j])  ==  sigmoid(s1[i]+s2[j]+b)
// One float4 per thread; q row (40 KB) lives in L2 across all 10000 rows;
// 400 MB output streamed with nontemporal stores (exceeds 192 MB L2 anyway).
// ---------------------------------------------------------------------------
__global__ __launch_bounds__(256) void denseatt_sigmoid_bcast(
    const float* __restrict__ pq,
    float* __restrict__ out)
{
    const int j4 = blockIdx.x * blockDim.x + threadIdx.x;  // float4 column index
    const int i  = blockIdx.y;
    if (j4 >= NR / 4) return;

    const float p = pq[i];                                  // uniform per row
    const v4f   q = *((const v4f*)(pq + NR) + j4);          // 16B aligned

    v4f r;
    r.x = __builtin_amdgcn_rcpf(1.0f + p * q.x);
    r.y = __builtin_amdgcn_rcpf(1.0f + p * q.y);
    r.z = __builtin_amdgcn_rcpf(1.0f + p * q.z);
    r.w = __builtin_amdgcn_rcpf(1.0f + p * q.w);

    v4f* dst = (v4f*)out + (size_t)i * (NR / 4) + j4;       // rows are 16B aligned
    __builtin_nontemporal_store(r, dst);
}

// ---------------------------------------------------------------------------
extern "C" void kernel_launch(void* const* d_in, const int* in_sizes, int n_in,
                              void* d_out, int out_size, void* d_ws, size_t ws_size,
                              hipStream_t stream) {
    const float* x = (const float*)d_in[0];   // (10000, 256)
    // d_in[1] = adj (10000x10000) -- unused by the reference computation
    const float* w = (const float*)d_in[2];   // (1, 512)
    const float* b = (const float*)d_in[3];   // (1,)
    float* out = (float*)d_out;               // (10000, 10000) f32
    float* pq  = (float*)d_ws;                // 2*NR floats = 80 KB scratch

    // Phase 1: tiny WMMA mat-vec + exp factorization (625 waves)
    denseatt_matvec_wmma<<<NR / 16, 32, 0, stream>>>(x, w, b, pq);

    // Phase 2: memory-bound broadcast sigmoid (store-limited, ~17 us floor)
    dim3 grid((NR / 4 + 255) / 256, NR);      // (10, 10000)
    denseatt_sigmoid_bcast<<<grid, 256, 0, stream>>>(pq, out);
}